// Prismatic_20323785245259
// MI455X (gfx1250) — compile-verified
//
#include <hip/hip_runtime.h>
#include <math.h>

typedef __attribute__((ext_vector_type(2))) float v2f;
typedef __attribute__((ext_vector_type(4))) float v4f;
typedef __attribute__((ext_vector_type(8))) float v8f;
typedef __attribute__((ext_vector_type(4))) int   i4;

#define HIDDEN 1024
#define DFF    4096
#define TOKENS 4096   // B*S = 2*2048

#if defined(__HIP_DEVICE_COMPILE__) && \
    __has_builtin(__builtin_amdgcn_global_load_async_to_lds_b128) && \
    __has_builtin(__builtin_amdgcn_s_wait_asynccnt)
#define USE_ASYNC_LDS 1
#else
#define USE_ASYNC_LDS 0
#endif

#define AS_GLOBAL __attribute__((address_space(1)))
#define AS_LDS    __attribute__((address_space(3)))

// ---------------- workspace layout (bytes) ----------------
static const size_t OFF_P0  = 0;                           // 16 KB  p0 per token
static const size_t OFF_THR = 16 * 1024;                   // 32 B   thr pairs x4 tensors
static const size_t OFF_SEL = 17 * 1024;                   // 16 B   radix-select state
static const size_t OFF_H1  = 32 * 1024;                   // 256 KB hi histogram
static const size_t OFF_H2  = 32 * 1024 + 256 * 1024;      // 512 KB lo histograms (x2)
static const size_t OFF_W1P = 1ull  * 1024 * 1024;         // 16 MB
static const size_t OFF_B1P = 17ull * 1024 * 1024;         // 16 KB
static const size_t OFF_W2P = 18ull * 1024 * 1024;         // 16 MB
static const size_t OFF_B2P = 34ull * 1024 * 1024;         // 4 KB
static const size_t OFF_ACT = 35ull * 1024 * 1024;         // 64 MB  gelu activations

__device__ __forceinline__ float gelu_tanh(float x) {
  const float c0 = 0.7978845608028654f; // sqrt(2/pi)
  float x3 = x * x * x;
  return 0.5f * x * (1.0f + tanhf(c0 * (x + 0.044715f * x3)));
}

// ---------------- router: LayerNorm + linear + softmax, keep p0 ----------------
__global__ __launch_bounds__(128) void router_kernel(
    const float* __restrict__ x, const float* __restrict__ gamma,
    const float* __restrict__ beta, const float* __restrict__ Wr,
    const float* __restrict__ br, float* __restrict__ p0)
{
  __shared__ float red[128];
  __shared__ float racc[128][8];
  const int t = threadIdx.x;
  const float* xr = x + (size_t)blockIdx.x * HIDDEN;

  float s = 0.f;
  for (int i = t; i < HIDDEN; i += 128) s += xr[i];
  red[t] = s; __syncthreads();
  for (int o = 64; o > 0; o >>= 1) { if (t < o) red[t] += red[t + o]; __syncthreads(); }
  float mu = red[0] * (1.0f / HIDDEN);
  __syncthreads();

  float v = 0.f;
  for (int i = t; i < HIDDEN; i += 128) { float d = xr[i] - mu; v += d * d; }
  red[t] = v; __syncthreads();
  for (int o = 64; o > 0; o >>= 1) { if (t < o) red[t] += red[t + o]; __syncthreads(); }
  float rstd = rsqrtf(red[0] * (1.0f / HIDDEN) + 1e-5f);
  __syncthreads();

  float acc[8];
#pragma unroll
  for (int e = 0; e < 8; ++e) acc[e] = 0.f;
  for (int i = t; i < HIDDEN; i += 128) {
    float h = (xr[i] - mu) * rstd * gamma[i] + beta[i];
#pragma unroll
    for (int e = 0; e < 8; ++e) acc[e] += h * Wr[i * 8 + e];
  }
#pragma unroll
  for (int e = 0; e < 8; ++e) racc[t][e] = acc[e];
  __syncthreads();
  for (int o = 64; o > 0; o >>= 1) {
    if (t < o) {
#pragma unroll
      for (int e = 0; e < 8; ++e) racc[t][e] += racc[t + o][e];
    }
    __syncthreads();
  }
  if (t == 0) {
    float lg[8], mx = -3.0e38f;
#pragma unroll
    for (int e = 0; e < 8; ++e) { lg[e] = racc[0][e] + br[e]; mx = fmaxf(mx, lg[e]); }
    float sum = 0.f, e0 = 0.f;
#pragma unroll
    for (int e = 0; e < 8; ++e) { float ex = expf(lg[e] - mx); sum += ex; if (e == 0) e0 = ex; }
    p0[blockIdx.x] = e0 / sum;
  }
}

// ---------------- exact rank selection via 2-pass radix histograms ----------------
__global__ void hist_hi_kernel(const float* __restrict__ w, int n, unsigned* __restrict__ hist) {
  for (int i = blockIdx.x * blockDim.x + threadIdx.x; i < n; i += gridDim.x * blockDim.x) {
    unsigned a = __float_as_uint(w[i]) & 0x7FFFFFFFu;
    atomicAdd(&hist[a >> 16], 1u);
  }
}

__global__ __launch_bounds__(256) void find_bin_kernel(
    const unsigned* __restrict__ hist, unsigned r0, unsigned r1, unsigned* __restrict__ sel)
{
  __shared__ unsigned part[256];
  __shared__ unsigned pfx[256];
  const int t = threadIdx.x;
  unsigned s = 0;
  for (int j = 0; j < 256; ++j) s += hist[t * 256 + j];
  part[t] = s;
  __syncthreads();
  if (t == 0) { unsigned cum = 0; for (int i = 0; i < 256; ++i) { pfx[i] = cum; cum += part[i]; } }
  __syncthreads();
  unsigned base = pfx[t], cnt = part[t];
  for (int which = 0; which < 2; ++which) {
    unsigned r = which ? r1 : r0;
    if (r >= base && r < base + cnt) {
      unsigned cum = base;
      for (int j = 0; j < 256; ++j) {
        unsigned h = hist[t * 256 + j];
        if (r < cum + h) { sel[which * 2 + 0] = (unsigned)(t * 256 + j); sel[which * 2 + 1] = r - cum; break; }
        cum += h;
      }
    }
  }
}

__global__ void hist_lo_kernel(const float* __restrict__ w, int n,
                               const unsigned* __restrict__ sel, unsigned* __restrict__ h2) {
  const unsigned hb0 = sel[0], hb1 = sel[2];
  for (int i = blockIdx.x * blockDim.x + threadIdx.x; i < n; i += gridDim.x * blockDim.x) {
    unsigned a = __float_as_uint(w[i]) & 0x7FFFFFFFu;
    unsigned hb = a >> 16;
    if (hb == hb0) atomicAdd(&h2[a & 0xFFFFu], 1u);
    if (hb == hb1) atomicAdd(&h2[65536u + (a & 0xFFFFu)], 1u);
  }
}

__global__ __launch_bounds__(256) void find_val_kernel(
    const unsigned* __restrict__ h2, const unsigned* __restrict__ sel, float* __restrict__ thr)
{
  __shared__ unsigned part[256];
  __shared__ unsigned pfx[256];
  const int t = threadIdx.x;
  for (int which = 0; which < 2; ++which) {
    const unsigned* h = h2 + (size_t)which * 65536u;
    unsigned hbin = sel[which * 2 + 0];
    unsigned r    = sel[which * 2 + 1];
    unsigned s = 0;
    for (int j = 0; j < 256; ++j) s += h[t * 256 + j];
    part[t] = s;
    __syncthreads();
    if (t == 0) { unsigned cum = 0; for (int i = 0; i < 256; ++i) { pfx[i] = cum; cum += part[i]; } }
    __syncthreads();
    unsigned base = pfx[t], cnt = part[t];
    if (r >= base && r < base + cnt) {
      unsigned cum = base;
      for (int j = 0; j < 256; ++j) {
        unsigned hc = h[t * 256 + j];
        if (r < cum + hc) { thr[which] = __uint_as_float((hbin << 16) | (unsigned)(t * 256 + j)); break; }
        cum += hc;
      }
    }
    __syncthreads();
  }
}

// w + 0.8*w*tier : top tier -> *0.2, bottom tier -> *1.8 (matches jnp.where order)
__global__ void perturb_kernel(const float* __restrict__ w, float* __restrict__ o,
                               int n, const float* __restrict__ thr) {
  const float tb = thr[0], tt = thr[1];
  for (int i = blockIdx.x * blockDim.x + threadIdx.x; i < n; i += gridDim.x * blockDim.x) {
    float v = w[i];
    float a = fabsf(v);
    float f = (a >= tt) ? 0.2f : ((a <= tb) ? 1.8f : 1.0f);
    o[i] = v * f;
  }
}

// ---------------- fp32 WMMA GEMM: C(MxN) = A(MxK) @ B(KxN), fused epilogues ----------------
// mode 0: C = gelu(A@B + bias[n])                     (write act)
// mode 1: C = p0[m]   * (A@B + bias[n])               (write out)
// mode 2: C += (1-p0[m]) * (A@B + bias[n])            (accumulate out)
__global__ __launch_bounds__(256) void gemm_wmma_f32(
    const float* __restrict__ A, const float* __restrict__ B,
    const float* __restrict__ bias, const float* __restrict__ p0,
    float* __restrict__ C, int M, int N, int K, int mode)
{
  __shared__ float As[128 * 20];   // 128 rows x 16 k, pad to 20 (16B-aligned float4 slots)
  __shared__ float Bs[8 * 288];    // 8 K-pairs x 128 cols interleaved: Bs[kp*288 + col*2 + (k&1)]
                                   // stride 288 == 32 (mod 64 banks): lane-halves complementary
  const int t    = threadIdx.x;
  const int lane = t & 31;
  const int wave = t >> 5;         // 8 wave32s
  const int wm   = wave & 3;       // 4 waves along M (32 rows each)
  const int wn   = wave >> 2;      // 2 waves along N (64 cols each)
  const int bm   = blockIdx.y * 128;
  const int bn   = blockIdx.x * 128;
  const int hi   = lane >> 4;      // lane half: K offset +2 (A/B), row offset +8 (C/D)
  const int lm   = lane & 15;

  const v8f zero = {0.f, 0.f, 0.f, 0.f, 0.f, 0.f, 0.f, 0.f};
  v8f c[2][4];
#pragma unroll
  for (int i = 0; i < 2; ++i)
#pragma unroll
    for (int j = 0; j < 4; ++j) c[i][j] = zero;

  for (int k0 = 0; k0 < K; k0 += 16) {
    // ---- A tile: 128x16 floats = 512 x 16B, contiguous per slot ----
#pragma unroll
    for (int r = 0; r < 2; ++r) {
      int idx = t + r * 256;
      int row = idx >> 2;
      int col = (idx & 3) << 2;
      const float* gsrc = A + (size_t)(bm + row) * K + k0 + col;
      float* ldst = &As[row * 20 + col];
#if USE_ASYNC_LDS
      // CDNA5 async DMA: global -> LDS, 16B per lane, tracked on ASYNCcnt
      __builtin_amdgcn_global_load_async_to_lds_b128(
          (AS_GLOBAL i4*)gsrc, (AS_LDS i4*)ldst, 0, 0);
#else
      *(v4f*)ldst = *(const v4f*)gsrc;
#endif
    }
    // ---- B tile: 16x128, scattered into K-pair-interleaved layout ----
#pragma unroll
    for (int r = 0; r < 2; ++r) {
      int idx = t + r * 256;
      int row = idx >> 5;          // k within chunk: 0..15
      int col = (idx & 31) << 2;   // 0..124
      v4f vb = *(const v4f*)(B + (size_t)(k0 + row) * N + bn + col);
      float* bp = &Bs[(row >> 1) * 288 + (row & 1)];
      bp[(col + 0) * 2] = vb.x;
      bp[(col + 1) * 2] = vb.y;
      bp[(col + 2) * 2] = vb.z;
      bp[(col + 3) * 2] = vb.w;
    }
#if USE_ASYNC_LDS
    __builtin_amdgcn_s_wait_asynccnt(0);
#endif
    __syncthreads();
#pragma unroll
    for (int kk = 0; kk < 4; ++kk) {        // four K=4 WMMA steps per chunk
      const int kb = kk * 4 + 2 * hi;       // K base for this lane half
      v2f a[2], b[4];
#pragma unroll
      for (int i = 0; i < 2; ++i) {
        int row = wm * 32 + i * 16 + lm;    // A frag: lanes0-15 K={kb,kb+1}, lanes16-31 K+2
        a[i] = *(const v2f*)(&As[row * 20 + kb]);
      }
#pragma unroll
      for (int j = 0; j < 4; ++j) {
        int colc = wn * 64 + j * 16 + lm;   // B frag: one b64 load, pair already adjacent
        b[j] = *(const v2f*)(&Bs[(kk * 2 + hi) * 288 + colc * 2]);
      }
#pragma unroll
      for (int i = 0; i < 2; ++i)
#pragma unroll
        for (int j = 0; j < 4; ++j)
          c[i][j] = __builtin_amdgcn_wmma_f32_16x16x4_f32(
              false, a[i], false, b[j], (short)0, c[i][j], false, false);
    }
    __syncthreads();
  }

#pragma unroll
  for (int j = 0; j < 4; ++j) {
    int colg = bn + wn * 64 + j * 16 + lm;
    float bv = bias[colg];
#pragma unroll
    for (int i = 0; i < 2; ++i) {
#pragma unroll
      for (int vv = 0; vv < 8; ++vv) {
        int rowg = bm + wm * 32 + i * 16 + vv + 8 * hi;   // C/D: VGPR v -> M = v (+8 for hi lanes)
        float acc = c[i][j][vv] + bv;
        size_t o = (size_t)rowg * N + colg;
        if (mode == 0)      C[o] = gelu_tanh(acc);
        else if (mode == 1) C[o] = p0[rowg] * acc;
        else                C[o] += (1.0f - p0[rowg]) * acc;
      }
    }
  }
}

// ---------------- launch ----------------
extern "C" void kernel_launch(void* const* d_in, const int* in_sizes, int n_in,
                              void* d_out, int out_size, void* d_ws, size_t ws_size,
                              hipStream_t stream)
{
  (void)in_sizes; (void)n_in; (void)out_size; (void)ws_size;
  const float* x   = (const float*)d_in[0];
  const float* W1  = (const float*)d_in[1];
  const float* b1  = (const float*)d_in[2];
  const float* W2  = (const float*)d_in[3];
  const float* b2  = (const float*)d_in[4];
  const float* gam = (const float*)d_in[5];
  const float* bet = (const float*)d_in[6];
  const float* Wr  = (const float*)d_in[7];
  const float* br  = (const float*)d_in[8];
  float* out = (float*)d_out;
  char* ws = (char*)d_ws;

  float*    p0  = (float*)(ws + OFF_P0);
  float*    thr = (float*)(ws + OFF_THR);
  unsigned* sel = (unsigned*)(ws + OFF_SEL);
  unsigned* h1  = (unsigned*)(ws + OFF_H1);
  unsigned* h2  = (unsigned*)(ws + OFF_H2);
  float* W1p = (float*)(ws + OFF_W1P);
  float* b1p = (float*)(ws + OFF_B1P);
  float* W2p = (float*)(ws + OFF_W2P);
  float* b2p = (float*)(ws + OFF_B2P);
  float* act = (float*)(ws + OFF_ACT);

  router_kernel<<<TOKENS, 128, 0, stream>>>(x, gam, bet, Wr, br, p0);

  const float* srcs[4] = {W1, b1, W2, b2};
  float*       dsts[4] = {W1p, b1p, W2p, b2p};
  const int    ns[4]   = {HIDDEN * DFF, DFF, DFF * HIDDEN, HIDDEN};
  for (int j = 0; j < 4; ++j) {
    int n = ns[j];
    int k = (int)((double)n * 0.1 / 2.0);   // matches Python int(n*sparsity/2)
    if (k < 1) k = 1;
    unsigned r_bot = (unsigned)(k - 1);     // s[k-1]
    unsigned r_top = (unsigned)(n - k);     // s[n-k]
    int blocks = (n + 255) / 256; if (blocks > 2048) blocks = 2048;
    float* thrj = thr + 2 * j;
    (void)hipMemsetAsync(h1, 0, 65536 * sizeof(unsigned), stream);
    hist_hi_kernel<<<blocks, 256, 0, stream>>>(srcs[j], n, h1);
    find_bin_kernel<<<1, 256, 0, stream>>>(h1, r_bot, r_top, sel);
    (void)hipMemsetAsync(h2, 0, 2 * 65536 * sizeof(unsigned), stream);
    hist_lo_kernel<<<blocks, 256, 0, stream>>>(srcs[j], n, sel, h2);
    find_val_kernel<<<1, 256, 0, stream>>>(h2, sel, thrj);
    perturb_kernel<<<blocks, 256, 0, stream>>>(srcs[j], dsts[j], n, thrj);
  }

  // clean expert
  gemm_wmma_f32<<<dim3(DFF / 128, TOKENS / 128), 256, 0, stream>>>(
      x, W1, b1, nullptr, act, TOKENS, DFF, HIDDEN, 0);
  gemm_wmma_f32<<<dim3(HIDDEN / 128, TOKENS / 128), 256, 0, stream>>>(
      act, W2, b2, p0, out, TOKENS, HIDDEN, DFF, 1);
  // perturbed expert (shared by experts 1..7), weighted by (1 - p0)
  gemm_wmma_f32<<<dim3(DFF / 128, TOKENS / 128), 256, 0, stream>>>(
      x, W1p, b1p, nullptr, act, TOKENS, DFF, HIDDEN, 0);
  gemm_wmma_f32<<<dim3(HIDDEN / 128, TOKENS / 128), 256, 0, stream>>>(
      act, W2p, b2p, p0, out, TOKENS, HIDDEN, DFF, 2);
}